// NMSHead_90108413870301
// MI455X (gfx1250) — compile-verified
//
#include <hip/hip_runtime.h>
#include <stdint.h>

typedef float        f4    __attribute__((ext_vector_type(4)));
typedef unsigned int u32x4 __attribute__((vector_size(16)));
typedef int          i32x4 __attribute__((vector_size(16)));
typedef int          i32x8 __attribute__((vector_size(32)));

#define IMG_H    1024
#define IMG_W    1024
#define TILE_H   16
#define RBUF     (TILE_H + 4)     // tile rows + 2-row halo top/bottom
#define NTHREADS 256              // 8 waves (wave32); 256 lanes * float4 = 1024 cols

__shared__ float s_in[RBUF * IMG_W];   // raw input rows        (80 KiB)
__shared__ float s_hm[RBUF * IMG_W];   // horizontal 5-max rows (80 KiB)

__device__ __forceinline__ f4 f4max(f4 a, f4 b) {
    f4 r;
    r.x = fmaxf(a.x, b.x);
    r.y = fmaxf(a.y, b.y);
    r.z = fmaxf(a.z, b.z);
    r.w = fmaxf(a.w, b.w);
    return r;
}

extern "C" __global__ __launch_bounds__(NTHREADS)
void nms_head_kernel(const float* __restrict__ in,      // (bs,1,H,W)
                     const float* __restrict__ scale,   // (bs,)
                     const float* __restrict__ center,  // (bs,2)
                     float* __restrict__ out,           // (bs,2,H,W) ++ (bs,H,W)
                     int bs)
{
    const int tid = threadIdx.x;
    const int img = blockIdx.y;
    const int r0  = blockIdx.x * TILE_H;
    const int c   = tid * 4;                 // this thread's 4-column group

    const float s  = scale[img];
    const float cx = center[img * 2 + 0];
    const float cy = center[img * 2 + 1];

    const size_t HW = (size_t)IMG_H * IMG_W;
    const float* in_img = in + (size_t)img * HW;

    // ---- Phase 1: one TDM descriptor stages the whole (RBUF x 1024) f32 tile into LDS.
    // Bottom halo: tensor_dim1 = IMG_H - start_row, so OOB rows are zero-filled by TDM.
    // Top halo (block 0 only): rows -2/-1 zero-filled with ds_stores below.
    int start_row = r0 - 2;
    int skip      = 0;
    if (start_row < 0) { skip = -start_row; start_row = 0; }   // uniform: only blockIdx.x==0
    const int nrows  = RBUF - skip;                            // tile rows to DMA
    const int nvalid = IMG_H - start_row;                      // rows before bottom OOB

    if (tid < 32) {                                            // wave 0 issues the DMA
        const uint32_t lds_byte = (uint32_t)(uintptr_t)&s_in[skip * IMG_W];
        const uint64_t ga       = (uint64_t)(uintptr_t)(in_img + (size_t)start_row * IMG_W);

        // D# group 0: count=1 | lds_addr | global_addr[56:0] | type=2
        u32x4 g0 = { 1u,
                     lds_byte,
                     (uint32_t)ga,
                     (uint32_t)((ga >> 32) & 0x01FFFFFFu) | 0x80000000u };

        // D# group 1: data_size=4B | tensor_dim0=1024 | tensor_dim1=nvalid |
        //             tile_dim0=1024 | tile_dim1=nrows | tensor_dim0_stride=1024
        i32x8 g1 = { (int)0x00020000u,                 // data_size=2 (4 bytes)
                     (int)0x04000000u,                 // tensor_dim0[15:0]=1024 @bits63:48
                     (int)((nvalid & 0xFFFF) << 16),   // tensor_dim1[15:0]   @bits95:80
                     (int)0x04000000u,                 // tile_dim0=1024      @bits127:112
                     (int)(nrows & 0xFFFF),            // tile_dim1           @bits143:128
                     (int)0x00000400u,                 // tensor_dim0_stride=1024 @bits191:160
                     0, 0 };

        i32x4 g2 = { 0, 0, 0, 0 };                     // unused (2D tile)
        i32x4 g3 = { 0, 0, 0, 0 };
        i32x8 g4 = { 0, 0, 0, 0, 0, 0, 0, 0 };         // extra operand on 6-arg toolchain: zeros

        __builtin_amdgcn_tensor_load_to_lds(g0, g1, g2, g3, g4, 0);
        __builtin_amdgcn_s_wait_tensorcnt(0);          // TDM writes to LDS complete
    }
    if (skip) {                                        // zero top-halo rows (block 0 only)
        *(f4*)&s_in[0 * IMG_W + c] = (f4)0.0f;
        *(f4*)&s_in[1 * IMG_W + c] = (f4)0.0f;
    }
    __syncthreads();

    // ---- Phase 2: horizontal 5-max per row (zero pad at column edges)
    for (int i = 0; i < RBUF; ++i) {
        const float* row = &s_in[i * IMG_W];
        f4 cur = *(const f4*)&row[c];
        float e0 = (tid > 0)            ? row[c - 2] : 0.0f;
        float e1 = (tid > 0)            ? row[c - 1] : 0.0f;
        float e6 = (tid < NTHREADS - 1) ? row[c + 4] : 0.0f;
        float e7 = (tid < NTHREADS - 1) ? row[c + 5] : 0.0f;
        float e2 = cur.x, e3 = cur.y, e4 = cur.z, e5 = cur.w;
        float m0 = fmaxf(e0, e1), m1 = fmaxf(e1, e2), m2 = fmaxf(e2, e3),
              m3 = fmaxf(e3, e4), m4 = fmaxf(e4, e5), m5 = fmaxf(e5, e6);
        f4 hm;
        hm.x = fmaxf(fmaxf(m0, m2), e4);   // max e0..e4
        hm.y = fmaxf(fmaxf(m1, m3), e5);   // max e1..e5
        hm.z = fmaxf(fmaxf(m2, m4), e6);   // max e2..e6
        hm.w = fmaxf(fmaxf(m3, m5), e7);   // max e3..e7
        *(f4*)&s_hm[i * IMG_W + c] = hm;
    }
    __syncthreads();

    // ---- Phase 3: vertical 5-max, mask, world coords, streaming stores
    float* out_wx = out + (size_t)(img * 2 + 0) * HW;
    float* out_wy = out + (size_t)(img * 2 + 1) * HW;
    float* out_m  = out + (size_t)(2 * bs) * HW + (size_t)img * HW;

    const float ub = (float)c - 0.5f * (float)IMG_W;
    f4 wx;
    wx.x = fmaf(ub + 0.0f, s, cx);
    wx.y = fmaf(ub + 1.0f, s, cx);
    wx.z = fmaf(ub + 2.0f, s, cx);
    wx.w = fmaf(ub + 3.0f, s, cx);

    for (int r = 0; r < TILE_H; ++r) {
        const int v = r0 + r;
        f4 h0 = *(const f4*)&s_hm[(r + 0) * IMG_W + c];
        f4 h1 = *(const f4*)&s_hm[(r + 1) * IMG_W + c];
        f4 h2 = *(const f4*)&s_hm[(r + 2) * IMG_W + c];
        f4 h3 = *(const f4*)&s_hm[(r + 3) * IMG_W + c];
        f4 h4 = *(const f4*)&s_hm[(r + 4) * IMG_W + c];
        f4 mx = f4max(f4max(f4max(h0, h1), f4max(h2, h3)), h4);
        f4 x  = *(const f4*)&s_in[(r + 2) * IMG_W + c];

        f4 m;
        m.x = (mx.x > 1e-5f && mx.x == x.x) ? 1.0f : 0.0f;
        m.y = (mx.y > 1e-5f && mx.y == x.y) ? 1.0f : 0.0f;
        m.z = (mx.z > 1e-5f && mx.z == x.z) ? 1.0f : 0.0f;
        m.w = (mx.w > 1e-5f && mx.w == x.w) ? 1.0f : 0.0f;

        const float wyv = fmaf(0.5f * (float)IMG_H - (float)v, s, cy);
        f4 owx = wx * m;
        f4 owy = wyv * m;

        const size_t off = (size_t)v * IMG_W + c;
        __builtin_nontemporal_store(owx, (f4*)&out_wx[off]);
        __builtin_nontemporal_store(owy, (f4*)&out_wy[off]);
        __builtin_nontemporal_store(m,   (f4*)&out_m[off]);
    }
}

extern "C" void kernel_launch(void* const* d_in, const int* in_sizes, int n_in,
                              void* d_out, int out_size, void* d_ws, size_t ws_size,
                              hipStream_t stream) {
    (void)n_in; (void)out_size; (void)d_ws; (void)ws_size;
    const float* in     = (const float*)d_in[0];   // input_map (bs,1,H,W) f32
    const float* scale  = (const float*)d_in[1];   // bev_scale (bs,)      f32
    const float* center = (const float*)d_in[2];   // bev_center (bs,2)    f32
    float* out = (float*)d_out;

    const int bs = in_sizes[1];                    // 16
    dim3 grid(IMG_H / TILE_H, bs);                 // 64 x 16 = 1024 blocks
    dim3 block(NTHREADS);
    hipLaunchKernelGGL(nms_head_kernel, grid, block, 0, stream,
                       in, scale, center, out, bs);
}